// Downsampler_24421184045082
// MI455X (gfx1250) — compile-verified
//
#include <hip/hip_runtime.h>
#include <hip/hip_bf16.h>

// Downsample: (16,8,1024,1024) f32, 4x4 Gaussian kernel, stride 4, VALID.
// Non-overlapping 4x4 windows -> each input byte read exactly once.
// Roofline: ~570 MB traffic / 23.3 TB/s => ~24.5us floor; 268 MFLOP is noise.
// => pure streaming kernel. CDNA5 async global->LDS (ASYNCcnt) path gives
// VGPR-free, deeply pipelined 512B/wave contiguous fetches.
//
// Key trick: the async op's 24-bit INST_OFFSET is added to BOTH the global
// address and the LDS address. LDS row stride (256*16B = 4096B) == global
// row stride (1024 f32 = 4096B), so all 4 row-fetches share one saddr/vaddr
// pair and differ only in offset:0/4096/8192/12288. GVS mode keeps the
// per-lane address math to a single v_lshlrev.

#define IN_W   1024
#define IN_H   1024
#define OUT_W  256
#define OUT_H  256
#define BLK    256          // one output row per block, 8 wave32s

__global__ __launch_bounds__(BLK) void Downsampler_24421184045082_kernel(
    const float* __restrict__ x,     // (128, 1024, 1024)
    const float* __restrict__ kern,  // (4, 4)
    float* __restrict__ out,         // (128, 256, 256)
    int total)
{
    // [row][thread][4 floats] : 16KB. Thread stride = 16B on both the async
    // writes and the ds_load_b128 readback -> conflict-free LDS banking.
    __shared__ __align__(16) float tile[4][BLK][4];

    const int tid     = threadIdx.x;
    const int row     = blockIdx.x;          // 0..32767 : one output row
    const int out_idx = row * BLK + tid;

    // Block-uniform (SALU): image and top input row of this output row.
    const int img = row >> 8;                // 0..127
    const int oy  = row & (OUT_H - 1);       // 0..255
    const unsigned long long sbase = (unsigned long long)(uintptr_t)
        (x + ((size_t)img * IN_H + (size_t)(oy * 4)) * IN_W);

    // Per-lane (VALU): byte offset of this thread's 16B segment in the row,
    // identical on the global side and the LDS side.
    const unsigned voff    = (unsigned)tid << 4;                       // tid*16
    const unsigned lds_off = (unsigned)(uintptr_t)(&tile[0][tid][0]);  // tid*16 (+base)

    // One clause: 4 async line fetches (global -> LDS), one per input row of
    // the 4x4 window. Per wave instruction: 32 lanes x 16B = 512B contiguous.
    // Tracked on ASYNCcnt; no VGPR return, no LOADcnt pressure.
    asm volatile(
        "s_clause 0x3\n\t"
        "global_load_async_to_lds_b128 %0, %1, %2 offset:0\n\t"
        "global_load_async_to_lds_b128 %0, %1, %2 offset:4096\n\t"
        "global_load_async_to_lds_b128 %0, %1, %2 offset:8192\n\t"
        "global_load_async_to_lds_b128 %0, %1, %2 offset:12288"
        :: "v"(lds_off), "v"(voff), "s"(sbase)
        : "memory");

    // Overlap: load the (uniform, cache-resident) 4x4 weights while the
    // async DMA is in flight.
    const float4 w0 = ((const float4*)kern)[0];
    const float4 w1 = ((const float4*)kern)[1];
    const float4 w2 = ((const float4*)kern)[2];
    const float4 w3 = ((const float4*)kern)[3];

    // Per-wave completion of our own LDS writes; threads read only their own
    // staged bytes, so no workgroup barrier is needed.
    asm volatile("s_wait_asynccnt 0x0" ::: "memory");

    const float4 v0 = *(const float4*)&tile[0][tid][0];
    const float4 v1 = *(const float4*)&tile[1][tid][0];
    const float4 v2 = *(const float4*)&tile[2][tid][0];
    const float4 v3 = *(const float4*)&tile[3][tid][0];

    float acc;
    acc  = v0.x * w0.x + v0.y * w0.y + v0.z * w0.z + v0.w * w0.w;
    acc += v1.x * w1.x + v1.y * w1.y + v1.z * w1.z + v1.w * w1.w;
    acc += v2.x * w2.x + v2.y * w2.y + v2.z * w2.z + v2.w * w2.w;
    acc += v3.x * w3.x + v3.y * w3.y + v3.z * w3.z + v3.w * w3.w;

    if (out_idx < total) {
        // Output is streamed once; keep it from thrashing L2.
        __builtin_nontemporal_store(acc, out + out_idx);
    }
}

extern "C" void kernel_launch(void* const* d_in, const int* in_sizes, int n_in,
                              void* d_out, int out_size, void* d_ws, size_t ws_size,
                              hipStream_t stream) {
    (void)in_sizes; (void)n_in; (void)d_ws; (void)ws_size;
    const float* x    = (const float*)d_in[0];   // 16*8*1024*1024 f32
    const float* kern = (const float*)d_in[1];   // 4*4 f32
    float* out        = (float*)d_out;           // 16*8*256*256 f32

    const int total  = out_size;                 // 8,388,608
    const int blocks = (total + BLK - 1) / BLK;  // 32,768 (exact)
    Downsampler_24421184045082_kernel<<<blocks, BLK, 0, stream>>>(x, kern, out, total);
}